// KrauseViTAttention_49143015801147
// MI455X (gfx1250) — compile-verified
//
#include <hip/hip_runtime.h>
#include <math.h>

typedef __attribute__((ext_vector_type(2))) float v2f;
typedef __attribute__((ext_vector_type(8))) float v8f;
typedef int v4i_g __attribute__((vector_size(16)));   // matches builtin's b128 type

#define NDIM 768
#define LQ 197
#define NH 12
#define DK 64
#define MTOT (32 * LQ)            // 6304
#define BHL (32 * NH * LQ)        // 75648
#define NKT (NDIM / 32)           // 24 K-chunks

// ---- CDNA5 async global->LDS support (guarded: falls back to sync path) ----
#if __has_builtin(__builtin_amdgcn_global_load_async_to_lds_b128) && \
    __has_builtin(__builtin_amdgcn_global_load_async_to_lds_b32)
#define HAVE_ASYNC_LDS 1
#else
#define HAVE_ASYNC_LDS 0
#endif

#if __has_builtin(__builtin_amdgcn_s_wait_asynccnt)
#define WAIT_ASYNC(n) __builtin_amdgcn_s_wait_asynccnt(n)
#else
#define WAIT_ASYNC(n) asm volatile("s_wait_asynccnt %0" ::"i"(n) : "memory")
#endif

__device__ __forceinline__ void wsync() {
    // wave-private LDS handoff: DS ops are in-order per wave; just stop the
    // compiler from reordering and pin convergence.
    asm volatile("" ::: "memory");
    __builtin_amdgcn_wave_barrier();
    asm volatile("" ::: "memory");
}

#if HAVE_ASYNC_LDS
__device__ __forceinline__ void async_copy_b128(const float* gp, float* lp) {
    __builtin_amdgcn_global_load_async_to_lds_b128(
        (__attribute__((address_space(1))) v4i_g*)gp,
        (__attribute__((address_space(3))) v4i_g*)lp, 0, 0);
}
__device__ __forceinline__ void async_copy_b32(const float* gp, float* lp) {
    __builtin_amdgcn_global_load_async_to_lds_b32(
        (__attribute__((address_space(1))) int*)gp,
        (__attribute__((address_space(3))) int*)lp, 0, 0);
}

__device__ __forceinline__ void prefetch_tile_async(
    const float* __restrict__ X, const float* __restrict__ W,
    float* xs, float* wt, int blockM, int blockN, int kb, int Mtot, int tid)
{
    // X tile: 128x32 floats, layout-preserving, b128 per lane (4 per thread)
    #pragma unroll
    for (int i = 0; i < 4; ++i) {
        int g = i * 256 + tid;
        int r = g >> 3;
        int c = (g & 7) << 2;
        int m = blockM + r;
        if (m >= Mtot) m = Mtot - 1;          // clamp (stores are guarded)
        async_copy_b128(X + (size_t)m * NDIM + kb + c, xs + r * 36 + c);
    }
    // W tile: 32x64 -> transposed scatter Wt[n][k], b32 per lane (8 per thread)
    #pragma unroll
    for (int i = 0; i < 8; ++i) {
        int g = i * 256 + tid;
        int k = g >> 6;
        int n = g & 63;
        async_copy_b32(W + (size_t)(kb + k) * NDIM + blockN + n, wt + n * 36 + k);
    }
}
#endif

// ---------------------------------------------------------------------------
// fp32 WMMA GEMM:  out[M,768] = X[M,768] @ W[768,768] + bias
// block = 256 threads (8 waves), tile 128(M) x 64(N), K staged 32 at a time,
// double-buffered LDS filled by async global->LDS DMA (ASYNCcnt).
// Each wave owns a 16(M) x 64(N) strip = 4 accumulators of 16x16.
// permute=1: scatter output into (B, H, L, dk) layout for Q/K/V.
// ---------------------------------------------------------------------------
__global__ __launch_bounds__(256)
void wmma_gemm_f32(const float* __restrict__ X, const float* __restrict__ W,
                   const float* __restrict__ bias, float* __restrict__ out,
                   int Mtot, int permute)
{
    __shared__ __align__(16) float Xs[2][128 * 36];  // [row][k], stride 36
    __shared__ __align__(16) float Wt[2][64 * 36];   // [n][k] transposed

    const int tid  = threadIdx.x;
    const int lane = tid & 31;
    const int wave = tid >> 5;
    const int l16  = lane & 15;
    const int kofs = (lane >> 4) << 1;          // lanes 16-31 hold K+2,K+3
    const int blockN = blockIdx.x * 64;
    const int blockM = blockIdx.y * 128;
    const int mrow = wave * 16 + l16;

    v8f acc[4] = {};

#if HAVE_ASYNC_LDS
    prefetch_tile_async(X, W, &Xs[0][0], &Wt[0][0], blockM, blockN, 0, Mtot, tid);
    for (int t = 0; t < NKT; ++t) {
        const int cur = t & 1;
        if (t + 1 < NKT) {
            prefetch_tile_async(X, W, &Xs[cur ^ 1][0], &Wt[cur ^ 1][0],
                                blockM, blockN, (t + 1) * 32, Mtot, tid);
            WAIT_ASYNC(12);   // 12 newer in flight -> current tile complete
        } else {
            WAIT_ASYNC(0);
        }
        __syncthreads();

        const float* xs = &Xs[cur][0];
        const float* wt = &Wt[cur][0];
        #pragma unroll
        for (int kk = 0; kk < 32; kk += 4) {
            v2f a = *(const v2f*)(&xs[mrow * 36 + kk + kofs]);
            #pragma unroll
            for (int u = 0; u < 4; ++u) {
                v2f b = *(const v2f*)(&wt[(u * 16 + l16) * 36 + kk + kofs]);
                acc[u] = __builtin_amdgcn_wmma_f32_16x16x4_f32(
                    false, a, false, b, (short)0, acc[u], false, false);
            }
        }
        __syncthreads();
    }
#else
    for (int kb = 0; kb < NDIM; kb += 32) {
        #pragma unroll
        for (int i = 0; i < 4; ++i) {
            int g = i * 256 + tid;
            int r = g >> 3;
            int c = (g & 7) << 2;
            float4 v = make_float4(0.f, 0.f, 0.f, 0.f);
            int m = blockM + r;
            if (m < Mtot) v = *(const float4*)(X + (size_t)m * NDIM + kb + c);
            *(float4*)(&Xs[0][r * 36 + c]) = v;
        }
        #pragma unroll
        for (int i = 0; i < 8; ++i) {
            int g = i * 256 + tid;
            int k = g >> 6;
            int n = g & 63;
            Wt[0][n * 36 + k] = W[(size_t)(kb + k) * NDIM + blockN + n];
        }
        __syncthreads();
        #pragma unroll
        for (int kk = 0; kk < 32; kk += 4) {
            v2f a = *(const v2f*)(&Xs[0][mrow * 36 + kk + kofs]);
            #pragma unroll
            for (int u = 0; u < 4; ++u) {
                v2f b = *(const v2f*)(&Wt[0][(u * 16 + l16) * 36 + kk + kofs]);
                acc[u] = __builtin_amdgcn_wmma_f32_16x16x4_f32(
                    false, a, false, b, (short)0, acc[u], false, false);
            }
        }
        __syncthreads();
    }
#endif

    // store: C/D layout — VGPR vi: lanes0-15 M=vi, lanes16-31 M=vi+8; N = lane&15
    const int hi = (lane >> 4) ? 8 : 0;
    #pragma unroll
    for (int u = 0; u < 4; ++u) {
        int n = blockN + u * 16 + l16;
        float bv = bias[n];
        #pragma unroll
        for (int vi = 0; vi < 8; ++vi) {
            int m = blockM + wave * 16 + hi + vi;
            if (m < Mtot) {
                float val = acc[u][vi] + bv;
                if (permute) {
                    int bI = m / LQ;
                    int l  = m - bI * LQ;
                    int h  = n >> 6;
                    int dd = n & 63;
                    out[((size_t)((bI * NH + h) * LQ + l) << 6) + dd] = val;
                } else {
                    out[(size_t)m * NDIM + n] = val;
                }
            }
        }
    }
}

// ---------------------------------------------------------------------------
// Gaussian-distance scores + locality mask + top-32 + softmax + weights*V.
// One wave per (b,h,i) row; 4 waves per block, wave-private LDS slices.
// ---------------------------------------------------------------------------
__global__ __launch_bounds__(128)
void gauss_topk_attn(const float* __restrict__ Qb, const float* __restrict__ Kb,
                     const float* __restrict__ Vb, const float* __restrict__ lsig,
                     float* __restrict__ wout, float* __restrict__ mout,
                     float* __restrict__ cons)
{
    __shared__ __align__(16) float qs_s[4][64];
    __shared__ float sc_s[4][200];   // scores (-inf invalid)
    __shared__ float fl_s[4][200];   // selection flags
    __shared__ float wk_s[4][200];   // work copy / normalized weights

    const int lane = threadIdx.x & 31;
    const int w    = threadIdx.x >> 5;
    const int row  = blockIdx.x * 4 + w;
    if (row >= BHL) return;

    const int b   = row / (NH * LQ);
    const int rem = row - b * (NH * LQ);
    const int h   = rem / LQ;
    const int i   = rem - h * LQ;
    const int bh  = b * NH + h;

    float* qs = qs_s[w];
    float* sc = sc_s[w];
    float* fl = fl_s[w];
    float* wk = wk_s[w];

    const float NEG_INF = -__builtin_inff();
    const float scale = -0.5f * expf(-2.0f * lsig[0]);

    // stage query row into LDS
    const float* qrow = Qb + ((size_t)(bh * LQ + i) << 6);
    for (int d = lane; d < DK; d += 32) qs[d] = qrow[d];
    wsync();

    const int ri = (i - 1) / 14, ci = (i - 1) % 14;  // only used when i>0
    const float* kbase = Kb + ((size_t)bh * LQ << 6);

    int cnt = 0;
    for (int j0 = 0; j0 < LQ; j0 += 32) {
        int j = j0 + lane;
        bool inr = (j < LQ);
        bool valid = false;
        if (inr) {
            valid = (i == 0) || (j == 0);
            if (!valid) {
                int rj = (j - 1) / 14, cj = (j - 1) % 14;
                int dr = ri - rj; if (dr < 0) dr = -dr;
                int dc = ci - cj; if (dc < 0) dc = -dc;
                valid = (dr <= 2) && (dc <= 2);
            }
        }
        float s = NEG_INF;
        if (valid) {
            const float* krow = kbase + ((size_t)j << 6);
            float acc = 0.f;
            #pragma unroll
            for (int d = 0; d < DK; d += 4) {
                float4 kv = *(const float4*)(krow + d);
                float4 qv = *(const float4*)(qs + d);
                float d0 = qv.x - kv.x, d1 = qv.y - kv.y;
                float d2 = qv.z - kv.z, d3 = qv.w - kv.w;
                acc += d0 * d0 + d1 * d1 + d2 * d2 + d3 * d3;
            }
            s = scale * acc;
        }
        if (inr) sc[j] = s;
        unsigned long long bal = __ballot(valid);
        cnt += (int)__popcll(bal);
    }
    wsync();

    // selection: patch rows have <=26 valid -> top-32 == all finite.
    if (cnt <= 32) {
        for (int j = lane; j < LQ; j += 32)
            fl[j] = (sc[j] != NEG_INF) ? 1.0f : 0.0f;
    } else {
        for (int j = lane; j < LQ; j += 32) { wk[j] = sc[j]; fl[j] = 0.0f; }
        wsync();
        for (int it = 0; it < 32; ++it) {
            float best = NEG_INF; int bidx = LQ;
            for (int j = lane; j < LQ; j += 32) {
                float v = wk[j];
                if (v > best) { best = v; bidx = j; }
            }
            #pragma unroll
            for (int off = 16; off > 0; off >>= 1) {
                float ov = __shfl_xor(best, off, 32);
                int   oi = __shfl_xor(bidx, off, 32);
                if (ov > best || (ov == best && oi < bidx)) { best = ov; bidx = oi; }
            }
            if (lane == 0) { fl[bidx] = 1.0f; wk[bidx] = NEG_INF; }
            wsync();
        }
    }
    wsync();

    // softmax over selected entries
    float m = NEG_INF;
    for (int j = lane; j < LQ; j += 32)
        if (fl[j] != 0.f) { float v = sc[j]; if (v > m) m = v; }
    #pragma unroll
    for (int off = 16; off > 0; off >>= 1) m = fmaxf(m, __shfl_xor(m, off, 32));

    float sum = 0.f;
    for (int j = lane; j < LQ; j += 32) {
        float ww = 0.f;
        if (fl[j] != 0.f) ww = expf(sc[j] - m);
        wk[j] = ww;
        sum += ww;
    }
    #pragma unroll
    for (int off = 16; off > 0; off >>= 1) sum += __shfl_xor(sum, off, 32);
    const float inv = 1.0f / sum;

    const size_t ro = (size_t)row * LQ;
    for (int j = lane; j < LQ; j += 32) {
        float ww = wk[j] * inv;
        wk[j] = ww;
        wout[ro + j] = ww;
        mout[ro + j] = (ww > 1e-6f) ? 1.0f : 0.0f;
    }
    wsync();

    // consensus = weights @ V  (sparse: only selected j, wave-uniform skip)
    const float* vbase = Vb + ((size_t)bh * LQ << 6);
    const int d = lane << 1;
    float c0 = 0.f, c1 = 0.f;
    for (int j = 0; j < LQ; ++j) {
        float ww = wk[j];              // uniform broadcast from LDS
        if (ww == 0.0f) continue;      // wave-uniform branch
        float2 vv = *(const float2*)(vbase + ((size_t)j << 6) + d);
        c0 += ww * vv.x;
        c1 += ww * vv.y;
    }
    float* cp = cons + ((size_t)(b * LQ + i) * NDIM + h * DK + d);
    *(float2*)cp = make_float2(c0, c1);
}

// ---------------------------------------------------------------------------
extern "C" void kernel_launch(void* const* d_in, const int* in_sizes, int n_in,
                              void* d_out, int out_size, void* d_ws, size_t ws_size,
                              hipStream_t stream) {
    const float* x    = (const float*)d_in[0];
    const float* Wq   = (const float*)d_in[1];
    const float* bq   = (const float*)d_in[2];
    const float* Wk   = (const float*)d_in[3];
    const float* bk   = (const float*)d_in[4];
    const float* Wv   = (const float*)d_in[5];
    const float* bv   = (const float*)d_in[6];
    const float* Wo   = (const float*)d_in[7];
    const float* bo   = (const float*)d_in[8];
    const float* lsig = (const float*)d_in[9];

    float* out  = (float*)d_out;                       // (B,L,D)   = 4,841,472
    float* wout = out + (size_t)32 * LQ * NDIM;        // (B,H,L,L) = 14,902,656
    float* mout = wout + (size_t)32 * NH * LQ * LQ;    // (B,H,L,L)

    const size_t QSZ = (size_t)32 * NH * LQ * DK;      // 4,841,472 floats
    float* Qb = (float*)d_ws;
    float* Kb = Qb + QSZ;
    float* Vb = Kb + QSZ;
    float* Cb = Vb + QSZ;

    dim3 gblk(256);
    dim3 ggrid(NDIM / 64, (MTOT + 127) / 128);         // (12, 50)

    wmma_gemm_f32<<<ggrid, gblk, 0, stream>>>(x, Wq, bq, Qb, MTOT, 1);
    wmma_gemm_f32<<<ggrid, gblk, 0, stream>>>(x, Wk, bk, Kb, MTOT, 1);
    wmma_gemm_f32<<<ggrid, gblk, 0, stream>>>(x, Wv, bv, Vb, MTOT, 1);

    gauss_topk_attn<<<dim3(BHL / 4), dim3(128), 0, stream>>>(
        Qb, Kb, Vb, lsig, wout, mout, Cb);

    wmma_gemm_f32<<<ggrid, gblk, 0, stream>>>(Cb, Wo, bo, out, MTOT, 0);
}